// Aggregation_4922032522023
// MI455X (gfx1250) — compile-verified
//
#include <hip/hip_runtime.h>

#define D_DIM 300
#define TILE  16

typedef __attribute__((ext_vector_type(2))) float v2f;
typedef __attribute__((ext_vector_type(8))) float v8f;

// ---------------------------------------------------------------------------
// Kernel 1: exclusive prefix sum of sizes[G] -> offsets[G]  (single block,
// 256 threads x 64 elements, 8-step Hillis-Steele scan of block partials).
// ---------------------------------------------------------------------------
__global__ void __launch_bounds__(256)
seg_offsets_kernel(const int* __restrict__ sizes, int* __restrict__ offsets, int G) {
    __shared__ int part[256];
    const int t     = threadIdx.x;
    const int per   = (G + 255) >> 8;
    const int start = t * per;

    int s = 0;
    for (int i = 0; i < per; ++i) {
        int idx = start + i;
        if (idx < G) s += sizes[idx];
    }
    part[t] = s;
    __syncthreads();

    // inclusive scan over the 256 partials (log2 steps)
    #pragma unroll
    for (int d = 1; d < 256; d <<= 1) {
        int v = (t >= d) ? part[t - d] : 0;
        __syncthreads();
        part[t] += v;
        __syncthreads();
    }
    int acc = part[t] - s;   // exclusive base for this thread's chunk

    for (int i = 0; i < per; ++i) {
        int idx = start + i;
        if (idx < G) { offsets[idx] = acc; acc += sizes[idx]; }
    }
}

// ---------------------------------------------------------------------------
// Kernel 2: one wave per (graph g, 16-column tile). Streams the segment's
// rows 4 at a time as the A matrix of V_WMMA_F32_16X16X4_F32 with B = ones,
// so the matrix unit does the fp32 column reduction.
//
// A layout (32-bit 16x4): lanes 0-15: M=lane,    VGPR0=K0, VGPR1=K1
//                         lanes 16-31: M=lane-16, VGPR0=K2, VGPR1=K3
// => lane serves column tile*16+(lane&15); lanes<16 load rows r,r+1,
//    lanes>=16 load rows r+2,r+3 (four 64B coalesced segments per chunk).
//
// Steady state inner loop: 2 NT loads (2nd folded as offset:1200) +
// 1 pointer increment + 1 WMMA. Masking only in the single tail chunk.
// ---------------------------------------------------------------------------
__global__ void __launch_bounds__(32)
segmean_wmma_kernel(const float* __restrict__ H,
                    const int*   __restrict__ sizes,
                    const int*   __restrict__ offsets,
                    float*       __restrict__ out) {
    const int g    = blockIdx.x;
    const int tile = blockIdx.y;
    const int lane = threadIdx.x;                 // 0..31, wave32

    const int n   = sizes[g];                     // wave-uniform
    const int off = offsets[g];                   // wave-uniform

    const int colRaw = tile * TILE + (lane & 15);
    const int col    = (colRaw < D_DIM) ? colRaw : (D_DIM - 1);   // clamp in-bounds
    const int rsel   = (lane >> 4) << 1;          // 0 for lanes 0-15, 2 for 16-31

    v8f c = {};                                   // fp32 accumulator (C matrix)
    v2f b; b.x = 1.0f; b.y = 1.0f;                // all-ones B (layout irrelevant)

    // ---- main loop: full 4-row chunks, no masking ----
    const int full = n & ~3;
    const float* p = H + (size_t)(off + rsel) * D_DIM + col;   // row r0 for this lane
    for (int r = 0; r < full; r += 4) {
        v2f a;
        a.x = __builtin_nontemporal_load(p);              // row r0
        a.y = __builtin_nontemporal_load(p + D_DIM);      // row r0+1 (imm offset 1200)
        c = __builtin_amdgcn_wmma_f32_16x16x4_f32(
                false, a, false, b, (short)0, c, false, false);
        p += 4 * D_DIM;
    }

    // ---- tail chunk: 1..3 remaining rows, value-masked (EXEC stays all-1s) ----
    if (full < n) {                               // uniform branch
        const int r0  = full + rsel;
        const int r1  = r0 + 1;
        const int rc0 = (r0 < n) ? r0 : (n - 1);  // clamped, in-bounds (n >= 1 here)
        const int rc1 = (r1 < n) ? r1 : (n - 1);
        float x0 = __builtin_nontemporal_load(H + (size_t)(off + rc0) * D_DIM + col);
        float x1 = __builtin_nontemporal_load(H + (size_t)(off + rc1) * D_DIM + col);
        v2f a;
        a.x = (r0 < n) ? x0 : 0.0f;
        a.y = (r1 < n) ? x1 : 0.0f;
        c = __builtin_amdgcn_wmma_f32_16x16x4_f32(
                false, a, false, b, (short)0, c, false, false);
    }

    // ---- extract column sums ----
    // D[m][*] identical across N. Value for M=m: (VGPR m, lanes 0-15) if m<8,
    // (VGPR m-8, lanes 16-31) if m>=8. 1KB LDS transpose -> coalesced store.
    __shared__ float sc[8][32];
#pragma unroll
    for (int i = 0; i < 8; ++i) sc[i][lane] = c[i];
    __syncthreads();

    if (lane < 16) {
        const float v    = (lane < 8) ? sc[lane][0] : sc[lane - 8][16];
        const int   colw = tile * TILE + lane;
        if (colw < D_DIM) {
            const float inv = 1.0f / fmaxf((float)n, 1.0f);   // zero-size -> 0
            out[(size_t)g * D_DIM + colw] = v * inv;
        }
    }
}

// ---------------------------------------------------------------------------
// Host-side launcher
// ---------------------------------------------------------------------------
extern "C" void kernel_launch(void* const* d_in, const int* in_sizes, int n_in,
                              void* d_out, int out_size, void* d_ws, size_t ws_size,
                              hipStream_t stream) {
    const float* H     = (const float*)d_in[0];   // [N, 300] fp32
    const int*   sizes = (const int*)d_in[1];     // [G] int32
    const int    G     = in_sizes[1];

    int*   offsets = (int*)d_ws;                  // G * 4 bytes of scratch
    float* out     = (float*)d_out;               // [G, 300] fp32

    seg_offsets_kernel<<<1, 256, 0, stream>>>(sizes, offsets, G);

    dim3 grid(G, (D_DIM + TILE - 1) / TILE);      // 16384 x 19 waves
    segmean_wmma_kernel<<<grid, 32, 0, stream>>>(H, sizes, offsets, out);
}